// CapsuleLayer_75746043232418
// MI455X (gfx1250) — compile-verified
//
#include <hip/hip_runtime.h>
#include <hip/hip_bf16.h>

// Problem constants (from reference setup_inputs)
#define BB   128   // batch
#define RR   4096  // route nodes
#define CIN  16    // in capsule dim
#define COUT 32    // out capsule dim
#define CC   2     // num capsules
#define EPSF 1e-12f

typedef __attribute__((ext_vector_type(2))) float v2f;
typedef __attribute__((ext_vector_type(8))) float v8f;

// ---------------------------------------------------------------------------
// Kernel 1: priors[c,b,r,o] = sum_i x[b,r,i] * W[c,r,i,o]
// Per (c,r): GEMM 128x16 @ 16x32 done with V_WMMA_F32_16X16X4_F32.
//   8 M-tiles (b), 2 N-tiles (o), 4 k-steps of K=4.
// Block = 128 threads (4 waves); each wave owns 2 M-tiles; RT=4 r's per block.
// ---------------------------------------------------------------------------
__global__ __launch_bounds__(128) void caps_priors_wmma(
    const float* __restrict__ x,       // [B,R,16]
    const float* __restrict__ w,       // [C,R,16,32]
    float* __restrict__ priors)        // [C,B,R,32]
{
    const int RT   = 4;
    const int c    = blockIdx.y;
    const int r0   = blockIdx.x * RT;
    const int wave = threadIdx.x >> 5;     // 0..3
    const int lane = threadIdx.x & 31;
    const int m    = lane & 15;            // row within tile (A/M) or col (B/N)
    const int kh   = lane >> 4;            // K-half select per ISA layout

    for (int rr = 0; rr < RT; ++rr) {
        const int r = r0 + rr;
        const float* wr = w + ((size_t)c * RR + r) * (CIN * COUT);

        v8f acc[2][2];
        const v8f vzero = {0.f,0.f,0.f,0.f,0.f,0.f,0.f,0.f};
        acc[0][0] = vzero; acc[0][1] = vzero; acc[1][0] = vzero; acc[1][1] = vzero;

        #pragma unroll
        for (int kk = 0; kk < 4; ++kk) {
            // A: 16x4 f32, lanes 0-15 hold {K0,K1}, lanes 16-31 hold {K2,K3}
            v2f afrag[2];
            #pragma unroll
            for (int t = 0; t < 2; ++t) {
                const int b = (2 * wave + t) * 16 + m;
                const float* xp = x + ((size_t)b * RR + r) * CIN + kk * 4 + 2 * kh;
                afrag[t] = *(const v2f*)xp;            // 8B aligned
            }
            // B: 4x16 f32, VGPR0: K = 2*kh, VGPR1: K = 2*kh+1, N = lane&15
            v2f bfrag[2];
            #pragma unroll
            for (int nt = 0; nt < 2; ++nt) {
                const int n    = nt * 16 + m;
                const int krow = kk * 4 + 2 * kh;
                bfrag[nt].x = wr[(size_t)krow * COUT + n];
                bfrag[nt].y = wr[(size_t)(krow + 1) * COUT + n];
            }
            #pragma unroll
            for (int t = 0; t < 2; ++t)
                #pragma unroll
                for (int nt = 0; nt < 2; ++nt)
                    acc[t][nt] = __builtin_amdgcn_wmma_f32_16x16x4_f32(
                        false, afrag[t], false, bfrag[nt],
                        (short)0, acc[t][nt], false, false);
        }

        // D layout: VGPR v -> row M = v + 8*kh; col N = lane&15
        #pragma unroll
        for (int t = 0; t < 2; ++t) {
            #pragma unroll
            for (int nt = 0; nt < 2; ++nt) {
                #pragma unroll
                for (int v = 0; v < 8; ++v) {
                    const int b = (2 * wave + t) * 16 + v + 8 * kh;
                    const int o = nt * 16 + m;
                    priors[(((size_t)c * BB + b) * RR + r) * COUT + o] = acc[t][nt][v];
                }
            }
        }
    }
}

// ---------------------------------------------------------------------------
// Kernel 2: iteration 0 — probs are uniform (softmax of zeros) so
// s = mean_r priors; outputs0 = squash(s).  One block per (c,b).
// ---------------------------------------------------------------------------
__global__ __launch_bounds__(256) void caps_iter0(
    const float* __restrict__ priors,  // [C,B,R,32]
    float* __restrict__ outbuf)        // [C,B,32]
{
    const int c = blockIdx.y, b = blockIdx.x, tid = threadIdx.x;
    const int o = tid & 31, g = tid >> 5;   // 8 r-groups
    const float* p = priors + ((size_t)c * BB + b) * RR * COUT;

    float part = 0.f;
    for (int r = g; r < RR; r += 8) part += p[(size_t)r * COUT + o];

    __shared__ float red[256];
    __shared__ float sArr[COUT];
    __shared__ float scaleSh;
    red[tid] = part;
    __syncthreads();
    if (tid < 32) {
        float s = 0.f;
        #pragma unroll
        for (int gg = 0; gg < 8; ++gg) s += red[gg * 32 + tid];
        sArr[tid] = s * (1.0f / RR);
    }
    __syncthreads();
    if (tid == 0) {
        float sn = 0.f;
        #pragma unroll
        for (int i = 0; i < COUT; ++i) sn += sArr[i] * sArr[i];
        scaleSh = (sn / (1.f + sn + EPSF)) * rsqrtf(sn + EPSF);
    }
    __syncthreads();
    if (tid < 32) outbuf[((size_t)c * BB + b) * COUT + tid] = sArr[tid] * scaleSh;
}

// ---------------------------------------------------------------------------
// Kernel 3: fused  delta(iter i) -> logits update -> softmax -> s -> squash
// (iter i+1).  One block per (c,b); logits held in 16 KB of LDS.
// ---------------------------------------------------------------------------
__global__ __launch_bounds__(256) void caps_route_iter(
    const float* __restrict__ priors,   // [C,B,R,32]
    float* __restrict__ logits,         // [C,B,R]
    const float* __restrict__ outPrev,  // [C,B,32]
    float* __restrict__ outNext,        // [C,B,32] (ws or d_out)
    int usePrevLogits, int storeLogits)
{
    const int c = blockIdx.y, b = blockIdx.x, tid = threadIdx.x;
    const float* p = priors + ((size_t)c * BB + b) * RR * COUT;
    float* lg = logits + ((size_t)c * BB + b) * RR;

    __shared__ float lgS[RR];          // 16 KB
    __shared__ float oPrev[COUT];
    __shared__ float red[256];
    __shared__ float zRed[8];
    __shared__ float sArr[COUT];
    __shared__ float scaleSh;

    if (tid < COUT) oPrev[tid] = outPrev[((size_t)c * BB + b) * COUT + tid];
    __syncthreads();

    // Pass 1: delta = <priors_r, outPrev>, logits update, running max
    float lmax = -3.402823466e38f;
    for (int r = tid; r < RR; r += 256) {
        const float* pr = p + (size_t)r * COUT;
        float d = 0.f;
        #pragma unroll
        for (int o = 0; o < COUT; ++o) d += pr[o] * oPrev[o];
        float v = d + (usePrevLogits ? lg[r] : 0.f);
        lgS[r] = v;
        lmax = fmaxf(lmax, v);
    }
    red[tid] = lmax;
    __syncthreads();
    #pragma unroll
    for (int s2 = 128; s2 > 0; s2 >>= 1) {
        if (tid < s2) red[tid] = fmaxf(red[tid], red[tid + s2]);
        __syncthreads();
    }
    const float mx = red[0];
    __syncthreads();

    // Pass 2: softmax-weighted sum over r
    const int o = tid & 31, g = tid >> 5;
    float spart = 0.f, zpart = 0.f;
    for (int r = g; r < RR; r += 8) {
        const float wgt = __expf(lgS[r] - mx);
        spart += wgt * p[(size_t)r * COUT + o];
        if (o == 0) zpart += wgt;
    }
    red[tid] = spart;
    if (o == 0) zRed[g] = zpart;
    __syncthreads();
    if (tid < 32) {
        float s = 0.f;
        #pragma unroll
        for (int gg = 0; gg < 8; ++gg) s += red[gg * 32 + tid];
        sArr[tid] = s;
    }
    __syncthreads();
    if (tid == 0) {
        float Z = 0.f;
        #pragma unroll
        for (int i = 0; i < 8; ++i) Z += zRed[i];
        const float invZ = 1.f / Z;
        float sn = 0.f;
        #pragma unroll
        for (int i = 0; i < COUT; ++i) {
            const float sv = sArr[i] * invZ;
            sArr[i] = sv;
            sn += sv * sv;
        }
        scaleSh = (sn / (1.f + sn + EPSF)) * rsqrtf(sn + EPSF);
    }
    __syncthreads();
    if (tid < 32) outNext[((size_t)c * BB + b) * COUT + tid] = sArr[tid] * scaleSh;
    if (storeLogits)
        for (int r = tid; r < RR; r += 256) lg[r] = lgS[r];
}

// ---------------------------------------------------------------------------
extern "C" void kernel_launch(void* const* d_in, const int* in_sizes, int n_in,
                              void* d_out, int out_size, void* d_ws, size_t ws_size,
                              hipStream_t stream) {
    (void)in_sizes; (void)n_in; (void)out_size; (void)ws_size;
    const float* x = (const float*)d_in[0];          // [128,4096,16] f32
    const float* w = (const float*)d_in[1];          // [2,4096,16,32] f32
    float* out = (float*)d_out;                      // [2,128,1,1,32] f32

    char* ws = (char*)d_ws;
    float* priors = (float*)ws;                                       // C*B*R*32 f32 (~134 MB)
    const size_t priorsBytes = (size_t)CC * BB * RR * COUT * sizeof(float);
    float* logitsBuf = (float*)(ws + priorsBytes);                    // C*B*R f32 (~4 MB)
    const size_t logitsBytes = (size_t)CC * BB * RR * sizeof(float);
    float* obuf = (float*)(ws + priorsBytes + logitsBytes);           // C*B*32 f32

    // Phase 1: WMMA GEMM -> priors
    dim3 gA(RR / 4, CC);
    caps_priors_wmma<<<gA, 128, 0, stream>>>(x, w, priors);

    // Phase 2: routing (3 iterations; iter boundaries fused)
    dim3 gB(BB, CC);
    caps_iter0<<<gB, 256, 0, stream>>>(priors, obuf);                                  // i=0
    caps_route_iter<<<gB, 256, 0, stream>>>(priors, logitsBuf, obuf, obuf, 0, 1);      // i=1
    caps_route_iter<<<gB, 256, 0, stream>>>(priors, logitsBuf, obuf, out, 1, 0);       // i=2
}